// IslandLossLayer_85358180040781
// MI455X (gfx1250) — compile-verified
//
#include <hip/hip_runtime.h>
#include <hip/hip_bf16.h>

typedef __attribute__((ext_vector_type(2))) float v2f;
typedef __attribute__((ext_vector_type(8))) float v8f;

#define C_CLS 7
#define D_DIM 1024
#define ALPHA 0.5f
#define LAMBDA2 0.5f

// ---------------------------------------------------------------------------
// K0: one block. Computes Gram matrix G (7x7) of centers, pair_dist scalar,
// and zeroes the classsum / counts scratch regions.
// ---------------------------------------------------------------------------
__global__ __launch_bounds__(256) void k_setup(const float* __restrict__ centers,
                                               float* __restrict__ ws_G,
                                               float* __restrict__ ws_pair,
                                               float* __restrict__ ws_counts,
                                               float* __restrict__ ws_classsum) {
    __shared__ float sG[C_CLS * C_CLS];
    const int t = threadIdx.x;

    if (t < C_CLS * C_CLS) {
        const int j = t / C_CLS, m = t % C_CLS;
        float s = 0.f;
        for (int d = 0; d < D_DIM; ++d)
            s += centers[j * D_DIM + d] * centers[m * D_DIM + d];
        sG[t] = s;
        ws_G[t] = s;
    }
    // zero scratch accumulators
    for (int i = t; i < 16 * D_DIM; i += 256) ws_classsum[i] = 0.f;
    if (t < 16) ws_counts[t] = 0.f;
    __syncthreads();

    if (t == 0) {
        float inv[C_CLS];
        for (int j = 0; j < C_CLS; ++j) inv[j] = 1.0f / sqrtf(sG[j * C_CLS + j]);
        float pd = 0.f;
        for (int j = 0; j < C_CLS; ++j)
            for (int m = 0; m < C_CLS; ++m)
                if (j != m) pd += sG[j * C_CLS + m] * inv[j] * inv[m] + 1.0f;
        ws_pair[0] = LAMBDA2 * pd;
    }
}

// ---------------------------------------------------------------------------
// K1: per-sample class index (labels are exact one-hot) + class counts.
// ---------------------------------------------------------------------------
__global__ __launch_bounds__(256) void k_cls(const float* __restrict__ labels,
                                             int* __restrict__ cls,
                                             float* __restrict__ ws_counts, int B) {
    __shared__ int cnt[C_CLS];
    const int t = threadIdx.x;
    const int r = blockIdx.x * 256 + t;
    if (t < C_CLS) cnt[t] = 0;
    __syncthreads();
    if (r < B) {
        int c = 0;
        float best = labels[(size_t)r * C_CLS];
#pragma unroll
        for (int m = 1; m < C_CLS; ++m) {
            float v = labels[(size_t)r * C_CLS + m];
            if (v > best) { best = v; c = m; }
        }
        cls[r] = c;
        atomicAdd(&cnt[c], 1);
    }
    __syncthreads();
    if (t < C_CLS) atomicAdd(&ws_counts[t], (float)cnt[t]);
}

// ---------------------------------------------------------------------------
// K2: classsum[16][1024] += labels.T @ feats via V_WMMA_F32_16X16X4_F32.
// grid.x = D/16 column tiles, grid.y = B/1024 K-chunks, 8 waves/block,
// each wave accumulates 128 samples = 32 fully-unrolled wmma k-steps (K=4).
//
// A (16x4 f32) lane layout (ISA 7.12.2): lanes 0-15: VGPR0=K0,VGPR1=K1;
// lanes 16-31: VGPR0=K2,VGPR1=K3; M = lane&15.  A[m][k] = (cls[s+k]==m).
// B (4x16 f32) mirrored: lanes 0-15 hold K0/K1, lanes 16-31 hold K2/K3,
// N = lane&15.  C/D: VGPR v, lane l -> (M = v + (l>=16?8:0), N = l&15).
// ---------------------------------------------------------------------------
__global__ __launch_bounds__(256) void k_classsum_wmma(const float* __restrict__ feats,
                                                       const int* __restrict__ cls,
                                                       float* __restrict__ ws_classsum,
                                                       int B, int D) {
    const int tid  = threadIdx.x;
    const int lane = tid & 31;
    const int wave = tid >> 5;
    const int col0 = blockIdx.x * 16;
    const int kbase = blockIdx.y * 1024 + wave * 128;

    const int m   = lane & 15;          // A row (class) this lane supplies
    const int n   = lane & 15;          // B/N column within tile
    const int klo = (lane >> 4) << 1;   // 0 for lanes 0-15, 2 for lanes 16-31

    // pre-computed streams; all per-step displacements are compile-time
    // constants so the unrolled loads use immediate offsets off one base.
    const float* __restrict__ fp0 = feats + (size_t)(kbase + klo) * D_DIM + col0 + n;
    const float* __restrict__ fp1 = fp0 + D_DIM;
    const int*   __restrict__ cp  = cls + kbase + klo;

    v8f acc = {};
#pragma unroll
    for (int it = 0; it < 32; ++it) {
        if (it < 28)  // compile-time: no branch after unrolling
            __builtin_prefetch(fp0 + it * 4 * D_DIM + 16 * D_DIM, 0, 3);
        const int c0 = cp[it * 4];
        const int c1 = cp[it * 4 + 1];
        v2f a, b;
        a.x = (c0 == m) ? 1.0f : 0.0f;
        a.y = (c1 == m) ? 1.0f : 0.0f;
        b.x = fp0[it * 4 * D_DIM];
        b.y = fp1[it * 4 * D_DIM];
        acc = __builtin_amdgcn_wmma_f32_16x16x4_f32(
            /*neg_a=*/false, a, /*neg_b=*/false, b,
            /*c_mod=*/(short)0, acc, /*reuse_a=*/false, /*reuse_b=*/false);
    }

    // reduce the 8 wave tiles in LDS, then one global f32 atomic per element
    __shared__ float red[8 * 256];
#pragma unroll
    for (int v = 0; v < 8; ++v) red[wave * 256 + lane * 8 + v] = acc[v];
    __syncthreads();

    float sum = 0.f;
#pragma unroll
    for (int w = 0; w < 8; ++w) sum += red[w * 256 + tid];
    const int l = tid >> 3, v = tid & 7;
    const int M = v + ((l >= 16) ? 8 : 0);
    const int N = l & 15;
    atomicAdd(&ws_classsum[M * D_DIM + col0 + N], sum);
}

// ---------------------------------------------------------------------------
// K3: result[i] = 0.5*||feats[i]-centers[cls_i]||^2 + pair_dist.
// Centers copied to LDS (28KB) via GLOBAL_LOAD_ASYNC_TO_LDS_B128 (ASYNCcnt),
// then one wave per row, float4 streaming, wave32 shuffle reduction.
// ---------------------------------------------------------------------------
__global__ __launch_bounds__(256) void k_result(const float* __restrict__ feats,
                                                const float* __restrict__ centers,
                                                const int* __restrict__ cls,
                                                const float* __restrict__ ws_pair,
                                                float* __restrict__ out, int B) {
    __shared__ float4 s4[C_CLS * (D_DIM / 4)];   // 1792 x 16B = 28KB
    const int t = threadIdx.x;

    // Async DMA centers -> LDS. Generic pointer low 32 bits = LDS byte offset.
    {
        const unsigned lds_base = (unsigned)(size_t)(void*)&s4[0];
#pragma unroll 2
        for (int i = t; i < C_CLS * (D_DIM / 4); i += 256) {
            const unsigned loff = lds_base + (unsigned)i * 16u;
            const void* gp = (const char*)centers + (size_t)i * 16u;
            asm volatile("global_load_async_to_lds_b128 %0, %1, off"
                         :: "v"(loff), "v"(gp) : "memory");
        }
        asm volatile("s_wait_asynccnt 0x0" ::: "memory");
    }
    __syncthreads();

    const float pair = ws_pair[0];
    const int lane = t & 31, wave = t >> 5;
    const int gw = blockIdx.x * 8 + wave;
    const int nw = gridDim.x * 8;
    const float4* f4 = (const float4*)feats;

    for (int r = gw; r < B; r += nw) {
        const int c = cls[r];
        const float4* frow = f4 + (size_t)r * (D_DIM / 4);
        const float4* crow = s4 + c * (D_DIM / 4);
        float acc = 0.f;
#pragma unroll
        for (int ch = 0; ch < 8; ++ch) {
            float4 f = frow[ch * 32 + lane];
            float4 cc = crow[ch * 32 + lane];
            float dx = f.x - cc.x, dy = f.y - cc.y, dz = f.z - cc.z, dw = f.w - cc.w;
            acc += dx * dx + dy * dy + dz * dz + dw * dw;
        }
#pragma unroll
        for (int off = 16; off > 0; off >>= 1) acc += __shfl_xor(acc, off, 32);
        if (lane == 0) out[r] = 0.5f * acc + pair;
    }
}

// ---------------------------------------------------------------------------
// K4: new_centers = centers - ALPHA*delta + (LAMBDA2/(C-1))*T  (C*D threads)
// ---------------------------------------------------------------------------
__global__ __launch_bounds__(256) void k_update(const float* __restrict__ centers,
                                                const float* __restrict__ ws_G,
                                                const float* __restrict__ ws_counts,
                                                const float* __restrict__ ws_classsum,
                                                float* __restrict__ out_nc) {
    const int idx = blockIdx.x * 256 + threadIdx.x;
    if (idx >= C_CLS * D_DIM) return;
    const int c = idx >> 10, d = idx & (D_DIM - 1);

    float inv[C_CLS], invsum = 0.f;
#pragma unroll
    for (int j = 0; j < C_CLS; ++j) {
        inv[j] = 1.0f / sqrtf(ws_G[j * C_CLS + j]);
        invsum += inv[j];
    }
    const float cnt = ws_counts[c];
    const float ctr = centers[idx];
    const float delta = (cnt * ctr - ws_classsum[c * D_DIM + d]) / (cnt + 1.0f);
    const float coef1 = inv[c] * (invsum - inv[c]);
    float t2 = 0.f;
#pragma unroll
    for (int m = 0; m < C_CLS; ++m) {
        if (m == c) continue;
        const float coef2 = ws_G[c * C_CLS + m] * inv[c] * (inv[m] * inv[m] * inv[m]);
        t2 += coef2 * centers[m * D_DIM + d];
    }
    const float T = ctr * coef1 - t2;
    out_nc[idx] = ctr - ALPHA * delta + (LAMBDA2 / (float)(C_CLS - 1)) * T;
}

// ---------------------------------------------------------------------------
extern "C" void kernel_launch(void* const* d_in, const int* in_sizes, int n_in,
                              void* d_out, int out_size, void* d_ws, size_t ws_size,
                              hipStream_t stream) {
    const float* feats   = (const float*)d_in[0];
    const float* labels  = (const float*)d_in[1];
    const float* centers = (const float*)d_in[2];
    float* out = (float*)d_out;                 // [0,B): result ; [B, B+C*D): new_centers

    const int B = in_sizes[1] / C_CLS;          // 32768
    const int D = in_sizes[0] / B;              // 1024

    // workspace carve (floats)
    float* ws          = (float*)d_ws;
    float* ws_pair     = ws;                    // 1
    float* ws_G        = ws + 8;                // 49
    float* ws_counts   = ws + 64;               // 16
    int*   ws_cls      = (int*)(ws + 128);      // B ints
    float* ws_classsum = ws + 128 + B;          // 16*1024 floats

    k_setup<<<1, 256, 0, stream>>>(centers, ws_G, ws_pair, ws_counts, ws_classsum);
    k_cls<<<(B + 255) / 256, 256, 0, stream>>>(labels, ws_cls, ws_counts, B);

    dim3 g1(D / 16, B / 1024);                  // 64 column tiles x 32 K-chunks
    k_classsum_wmma<<<g1, 256, 0, stream>>>(feats, ws_cls, ws_classsum, B, D);

    k_result<<<256, 256, 0, stream>>>(feats, centers, ws_cls, ws_pair, out, B);
    k_update<<<(C_CLS * D_DIM + 255) / 256, 256, 0, stream>>>(centers, ws_G, ws_counts,
                                                              ws_classsum, out + B);
}